// GCN_6270652252215
// MI455X (gfx1250) — compile-verified
//
#include <hip/hip_runtime.h>
#include <hip/hip_bf16.h>
#include <math.h>

#define NN   50000
#define EE   800000
#define FIN  128
#define HH   96
#define CC   16
#define GG   128

static_assert(NN % 16 == 0, "N must be multiple of 16 for WMMA tiling");
static_assert(HH % 4 == 0 && FIN % 4 == 0, "feature dims must be vec4-able");

typedef float v2f __attribute__((ext_vector_type(2)));
typedef float v8f __attribute__((ext_vector_type(8)));

__device__ __forceinline__ float gelu_exact(float x) {
    return 0.5f * x * (1.0f + erff(x * 0.7071067811865475f));
}

// ---------------------------------------------------------------- degree / norm
__global__ void k_init_deg(float* deg) {
    int i = blockIdx.x * blockDim.x + threadIdx.x;
    if (i < NN) deg[i] = 2.0f;               // improved=True self loop weight 2
}

__global__ void k_count(const int* __restrict__ dst, float* deg) {
    int e = blockIdx.x * blockDim.x + threadIdx.x;
    if (e < EE) atomicAdd(&deg[dst[e]], 1.0f);
}

__global__ void k_finish_deg(float* dis, float* selfc) {
    int i = blockIdx.x * blockDim.x + threadIdx.x;
    if (i < NN) {
        float d = dis[i];                    // deg (incl. +2)
        dis[i]   = rsqrtf(d);
        selfc[i] = 2.0f / d;                 // 2 * dis^2
    }
}

__global__ void k_edge_norm(const int* __restrict__ src, const int* __restrict__ dst,
                            const float* __restrict__ dis, float* __restrict__ nrm) {
    int e = blockIdx.x * blockDim.x + threadIdx.x;
    if (e < EE) nrm[e] = dis[src[e]] * dis[dst[e]];
}

// ---------------------------------------------------------------- WMMA GEMM
// Out[N x HH] = A[N x K] @ W[K x HH].  One block = 16 rows, 6 wave32 = 6 col tiles.
// A tile staged into LDS with the CDNA5 async global->LDS path.
__global__ void __launch_bounds__(192)
k_gemm_wmma(const float* __restrict__ A, const float* __restrict__ W,
            float* __restrict__ Out, int K) {
    // stride K+4: rows stay 16B aligned for B128, and the 16 column readers
    // land on banks m*4 (mod 64) -> 16 distinct banks, conflict free.
    __shared__ __align__(16) float As[16 * (FIN + 4)];
    const int row0   = blockIdx.x * 16;
    const int tid    = threadIdx.x;
    const int stride = K + 4;
    const int Kq     = K >> 2;               // float4 chunks per row

    // LDS byte address of As: low 32 bits of the generic pointer ARE the DS byte
    // offset (ISA Sec.10.2 aperture mapping).  The ptrtoint also forces `As` to
    // escape, so the memory-clobbering asm below is a legal writer of it.
    const unsigned lds_base = (unsigned)(size_t)(const void*)As;

    // async stage of the 16 x K A tile into LDS
    for (int v = tid; v < 16 * Kq; v += 192) {
        int r = v / Kq, c = (v - r * Kq) << 2;
        const float* gp = A + (size_t)(row0 + r) * K + c;
        unsigned lds_byte = lds_base + (unsigned)((r * stride + c) * 4);
        asm volatile("global_load_async_to_lds_b128 %0, %1, off"
                     :: "v"(lds_byte), "v"(gp) : "memory");
    }
    asm volatile("s_wait_asynccnt 0x0" ::: "memory");
    __syncthreads();

    const int wave = tid >> 5;
    const int lane = tid & 31;
    const int m    = lane & 15;              // row (A) / col (B) within tile
    const int kh   = (lane >> 4) << 1;       // lanes 0-15 -> {k,k+1}, 16-31 -> {k+2,k+3}
    const int col0 = wave * 16;

    v8f acc = {};
    for (int k = 0; k < K; k += 4) {
        v2f a, b;
        a.x = As[m * stride + k + kh];
        a.y = As[m * stride + k + kh + 1];
        b.x = W[(k + kh)     * HH + col0 + m];
        b.y = W[(k + kh + 1) * HH + col0 + m];
        acc = __builtin_amdgcn_wmma_f32_16x16x4_f32(
            /*neg_a=*/false, a, /*neg_b=*/false, b,
            /*c_mod=*/(short)0, acc, /*reuse_a=*/false, /*reuse_b=*/false);
    }

    // C/D layout: VGPR r -> M = r + (lane<16 ? 0 : 8), N = lane&15
    const int rbase = row0 + ((lane >> 4) << 3);
    const int col   = col0 + m;
#pragma unroll
    for (int r = 0; r < 8; r++)
        Out[(rbase + r) * HH + col] = acc[r];
}

// ---------------------------------------------------------------- conv pieces
__global__ void k_self_init4(const float4* __restrict__ xw4, const float* __restrict__ selfc,
                             const float* __restrict__ bias, float4* __restrict__ out4) {
    int idx = blockIdx.x * blockDim.x + threadIdx.x;
    if (idx < NN * HH / 4) {
        int p = idx * 4;
        int i = p / HH, f = p - i * HH;
        float  sc = selfc[i];
        float4 v  = xw4[idx];
        float4 b  = *(const float4*)(bias + f);
        out4[idx] = make_float4(b.x + sc * v.x, b.y + sc * v.y,
                                b.z + sc * v.z, b.w + sc * v.w);
    }
}

// 4 features per thread: one b128 gather + 4 f32 atomics; 4x fewer index loads.
__global__ void k_scatter(const int* __restrict__ src, const int* __restrict__ dst,
                          const float* __restrict__ nrm, const float* __restrict__ xw,
                          float* __restrict__ out) {
    const int Q   = HH / 4;                  // 24 quads per edge
    int idx = blockIdx.x * blockDim.x + threadIdx.x;
    if (idx < EE * Q) {
        int e = idx / Q;
        int f = (idx - e * Q) << 2;
        int s = src[e], d = dst[e];
        float n = nrm[e];
        if (f == 0) {                        // prefetch an upcoming edge's source row
            int ep = e + 128;
            if (ep < EE) __builtin_prefetch(xw + (size_t)src[ep] * HH, 0, 0);
        }
        float4 v = *(const float4*)(xw + (size_t)s * HH + f);
        float* o = out + (size_t)d * HH + f;
        atomicAdd(o + 0, n * v.x);
        atomicAdd(o + 1, n * v.y);
        atomicAdd(o + 2, n * v.z);
        atomicAdd(o + 3, n * v.w);
    }
}

__global__ void k_gelu4(const float4* __restrict__ in4, float4* __restrict__ out4) {
    int idx = blockIdx.x * blockDim.x + threadIdx.x;
    if (idx < NN * HH / 4) {
        float4 v = in4[idx];
        out4[idx] = make_float4(gelu_exact(v.x), gelu_exact(v.y),
                                gelu_exact(v.z), gelu_exact(v.w));
    }
}

__global__ void k_residual4(float4* __restrict__ h4, const float4* __restrict__ t4) {
    int idx = blockIdx.x * blockDim.x + threadIdx.x;
    if (idx < NN * HH / 4) {
        float4 a = h4[idx], b = t4[idx];
        h4[idx] = make_float4(a.x + b.x, a.y + b.y, a.z + b.z, a.w + b.w);
    }
}

// ---------------------------------------------------------------- pooling + head
__device__ __forceinline__ int lower_bound_i(const int* b, int n, int key) {
    int lo = 0, hi = n;
    while (lo < hi) { int mid = (lo + hi) >> 1; if (b[mid] < key) lo = mid + 1; else hi = mid; }
    return lo;
}

__global__ void k_pool(const float* __restrict__ h, const int* __restrict__ batch,
                       float* __restrict__ p) {
    int g = blockIdx.x;          // 0..GG-1
    int f = threadIdx.x;         // 0..HH-1
    int s = lower_bound_i(batch, NN, g);
    int e = lower_bound_i(batch, NN, g + 1);
    float sum = 0.0f, mx = -INFINITY;
    for (int i = s; i < e; i++) {
        float v = h[(size_t)i * HH + f];
        sum += v;
        mx = fmaxf(mx, v);
    }
    int cnt = e - s;
    p[g * (2 * HH) + f]      = cnt ? sum / (float)cnt : 0.0f;
    p[g * (2 * HH) + HH + f] = cnt ? mx : 0.0f;
}

__global__ void k_mlp1(const float* __restrict__ p, const float* __restrict__ Wc1,
                       const float* __restrict__ bc1, float* __restrict__ p1) {
    int g = blockIdx.x, j = threadIdx.x;     // GG x HH
    float acc = bc1[j];
    for (int k = 0; k < 2 * HH; k++) acc += p[g * (2 * HH) + k] * Wc1[k * HH + j];
    p1[g * HH + j] = gelu_exact(acc);
}

__global__ void k_mlp2(const float* __restrict__ p1, const float* __restrict__ Wc2,
                       const float* __restrict__ bc2, float* __restrict__ out) {
    int idx = blockIdx.x * blockDim.x + threadIdx.x;
    if (idx < GG * CC) {
        int g = idx / CC, c = idx - g * CC;
        float acc = bc2[c];
        for (int j = 0; j < HH; j++) acc += p1[g * HH + j] * Wc2[j * CC + c];
        out[idx] = acc;
    }
}

// ---------------------------------------------------------------- launch
extern "C" void kernel_launch(void* const* d_in, const int* in_sizes, int n_in,
                              void* d_out, int out_size, void* d_ws, size_t ws_size,
                              hipStream_t stream) {
    const float* x    = (const float*)d_in[0];
    const int*   ei   = (const int*)  d_in[1];   // [2, E]
    const int*   bat  = (const int*)  d_in[2];
    const float* W0   = (const float*)d_in[3];
    const float* b0   = (const float*)d_in[4];
    const float* W1   = (const float*)d_in[5];
    const float* b1   = (const float*)d_in[6];
    const float* W2   = (const float*)d_in[7];
    const float* b2   = (const float*)d_in[8];
    const float* Wc1  = (const float*)d_in[9];
    const float* bc1  = (const float*)d_in[10];
    const float* Wc2  = (const float*)d_in[11];
    const float* bc2  = (const float*)d_in[12];
    const int* src = ei;
    const int* dst = ei + EE;
    float* out = (float*)d_out;

    // workspace layout
    char*  w   = (char*)d_ws;
    size_t off = 0;
    auto alloc = [&](size_t bytes) -> float* {
        float* p = (float*)(w + off);
        off += (bytes + 255) & ~(size_t)255;
        return p;
    };
    float* dis   = alloc(NN * sizeof(float));
    float* selfc = alloc(NN * sizeof(float));
    float* nrm   = alloc(EE * sizeof(float));
    float* u     = alloc((size_t)NN * HH * sizeof(float));   // X @ W
    float* t     = alloc((size_t)NN * HH * sizeof(float));   // gelu buf / conv accum
    float* h     = alloc((size_t)NN * HH * sizeof(float));   // running hidden state
    float* p     = alloc((size_t)GG * 2 * HH * sizeof(float));
    float* p1    = alloc((size_t)GG * HH * sizeof(float));
    (void)ws_size; (void)in_sizes; (void)n_in; (void)out_size;

    const int TB = 256;
    const int NH4 = NN * HH / 4;
    const int EQ  = EE * (HH / 4);
    dim3 blkN((NN + TB - 1) / TB), blkE((EE + TB - 1) / TB);
    dim3 blkNH4((NH4 + TB - 1) / TB), blkEQ((EQ + TB - 1) / TB);

    // degrees / norms (computed once, reused by all 3 layers)
    k_init_deg  <<<blkN, TB, 0, stream>>>(dis);
    k_count     <<<blkE, TB, 0, stream>>>(dst, dis);
    k_finish_deg<<<blkN, TB, 0, stream>>>(dis, selfc);
    k_edge_norm <<<blkE, TB, 0, stream>>>(src, dst, dis, nrm);

    // layer 0: GCNConv(128 -> 96)  -> h
    k_gemm_wmma <<<NN / 16, 192, 0, stream>>>(x, W0, u, FIN);
    k_self_init4<<<blkNH4, TB, 0, stream>>>((const float4*)u, selfc, b0, (float4*)h);
    k_scatter   <<<blkEQ,  TB, 0, stream>>>(src, dst, nrm, u, h);

    // layer 1: h += GCNConv(gelu(h))
    k_gelu4     <<<blkNH4, TB, 0, stream>>>((const float4*)h, (float4*)t);
    k_gemm_wmma <<<NN / 16, 192, 0, stream>>>(t, W1, u, HH);
    k_self_init4<<<blkNH4, TB, 0, stream>>>((const float4*)u, selfc, b1, (float4*)t);
    k_scatter   <<<blkEQ,  TB, 0, stream>>>(src, dst, nrm, u, t);
    k_residual4 <<<blkNH4, TB, 0, stream>>>((float4*)h, (const float4*)t);

    // layer 2: h += GCNConv(gelu(h))
    k_gelu4     <<<blkNH4, TB, 0, stream>>>((const float4*)h, (float4*)t);
    k_gemm_wmma <<<NN / 16, 192, 0, stream>>>(t, W2, u, HH);
    k_self_init4<<<blkNH4, TB, 0, stream>>>((const float4*)u, selfc, b2, (float4*)t);
    k_scatter   <<<blkEQ,  TB, 0, stream>>>(src, dst, nrm, u, t);
    k_residual4 <<<blkNH4, TB, 0, stream>>>((float4*)h, (const float4*)t);

    // final gelu + pooling + MLP head
    k_gelu4<<<blkNH4, TB, 0, stream>>>((const float4*)h, (float4*)t);
    k_pool<<<GG, HH, 0, stream>>>(t, bat, p);
    k_mlp1<<<GG, HH, 0, stream>>>(p, Wc1, bc1, p1);
    k_mlp2<<<(GG * CC + TB - 1) / TB, TB, 0, stream>>>(p1, Wc2, bc2, out);
}